// AAMSoftmax_15006615733568
// MI455X (gfx1250) — compile-verified
//
#include <hip/hip_runtime.h>
#include <math.h>
#include <stdint.h>

typedef __attribute__((ext_vector_type(2))) float v2f;
typedef __attribute__((ext_vector_type(4))) float v4f;
typedef __attribute__((ext_vector_type(8))) float v8f;

#define BB      512
#define DD      256
#define CC      100000
#define NBLK    782            // ceil(100000 / 128) column blocks
#define NSTRIP  (NBLK * 2)     // 64-wide column strips (1564)
#define SCALE_F 30.0f
#define MARGIN_F 0.2f

// K-chunking / LDS layout for the GEMM kernel
#define KC      32             // K chunk
#define SS      36             // LDS row stride in floats: 36 mod 64 -> conflict-free
                               // b64 frag reads; 36*4 = 144 B -> rows 16B aligned
#define A_FL    (64 * SS)      // 2304 floats per A buffer
#define B_FL    (128 * SS)     // 4608 floats per B buffer
#define BUF_FL  (A_FL + B_FL)  // 6912 floats per double-buffer slot
#define SMEM_FL (2 * BUF_FL)   // 13824 floats = 54 KB LDS

// workspace layout (float offsets)
#define OFF_RNORM 0
#define OFF_PMAX  (OFF_RNORM + CC)
#define OFF_PSUM  (OFF_PMAX + BB * NSTRIP)
#define OFF_TLOG  (OFF_PSUM + BB * NSTRIP)

// ---------------------------------------------------------------------------
// Kernel 1: per-row inverse L2 norm of weight. One wave32 per row.
// ---------------------------------------------------------------------------
__global__ __launch_bounds__(256) void wnorm_kernel(const float* __restrict__ w,
                                                    float* __restrict__ rnorm) {
  const int wave = threadIdx.x >> 5;
  const int lane = threadIdx.x & 31;
  const int row  = blockIdx.x * 8 + wave;
  const float* p = w + (size_t)row * DD + lane * 8;
  v4f a = *(const v4f*)p;
  v4f b = *(const v4f*)(p + 4);
  float s = a[0]*a[0] + a[1]*a[1] + a[2]*a[2] + a[3]*a[3]
          + b[0]*b[0] + b[1]*b[1] + b[2]*b[2] + b[3]*b[3];
#pragma unroll
  for (int mk = 16; mk >= 1; mk >>= 1) s += __shfl_xor(s, mk, 32);
  if (lane == 0) rnorm[row] = rsqrtf(s);
}

// ---------------------------------------------------------------------------
// Kernel 2: fused GEMM + margin + online softmax partials.
// Workgroup = 8 waves, 4(M)x2(N); wave tile = 16 rows x 64 cols; WG tile 64x128.
// Data path: coalesced GLOBAL_LOAD_ASYNC_TO_LDS_B128 (ASYNCcnt, double
// buffered over K chunks of 32) -> conflict-free ds_load_b64 fragment reads
// -> V_WMMA_F32_16X16X4_F32.
// ---------------------------------------------------------------------------
__global__ __launch_bounds__(256) void gemm_softmax_kernel(
    const float* __restrict__ emb, const float* __restrict__ wgt,
    const int* __restrict__ labels, float* __restrict__ ws) {
  __shared__ float smem[SMEM_FL];

  float* rnorm = ws + OFF_RNORM;
  float* pmax  = ws + OFF_PMAX;
  float* psum  = ws + OFF_PSUM;
  float* tlog  = ws + OFF_TLOG;

  const float NEG_INF = -__builtin_inff();

  const int tid   = threadIdx.x;
  const int wave  = tid >> 5;
  const int lane  = tid & 31;
  const int half  = lane >> 4;     // 0: lanes 0-15, 1: lanes 16-31
  const int lm    = lane & 15;
  const int waveM = wave & 3;
  const int waveN = wave >> 2;
  const int rowB0 = blockIdx.y * 64;        // workgroup row base
  const int colB0 = blockIdx.x * 128;       // workgroup col base
  const int row0  = rowB0 + waveM * 16;     // wave row base
  const int col0  = colB0 + waveN * 64;     // wave col base

  // ---- per-thread async-copy descriptors (16B per lane per instruction) ----
  // A chunk: 32 rows x 64 k = 512 float4 -> 2 per thread.
  // B chunk: 64 rows x 64 k = 1024 float4 -> 4 per thread.
  const uint32_t smem_base = (uint32_t)(uintptr_t)smem;  // LDS byte offset
  uint32_t a_l[2]; const float* a_g[2];
#pragma unroll
  for (int i = 0; i < 2; i++) {
    int q = i * 256 + tid, r = q >> 4, kk = (q & 15) * 4;
    a_l[i] = smem_base + (uint32_t)((r * SS + kk) * 4);
    a_g[i] = emb + (size_t)(rowB0 + r) * DD + kk;
  }
  uint32_t b_l[4]; const float* b_g[4];
#pragma unroll
  for (int i = 0; i < 4; i++) {
    int q = i * 256 + tid, r = q >> 4, kk = (q & 15) * 4;
    int c = colB0 + r;
    int cc = (c < CC) ? c : (CC - 1);       // clamp OOB columns (masked later)
    b_l[i] = smem_base + (uint32_t)((A_FL + r * SS + kk) * 4);
    b_g[i] = wgt + (size_t)cc * DD + kk;
  }

  // ---- fragment read offsets in LDS (floats) ----
  const int aoff = (waveM * 16 + lm) * SS + 2 * half;
  int boff[4];
#pragma unroll
  for (int t = 0; t < 4; t++)
    boff[t] = A_FL + (waveN * 64 + t * 16 + lm) * SS + 2 * half;

  // column bookkeeping for the epilogue
  int colc[4];
#pragma unroll
  for (int t = 0; t < 4; t++) {
    int c = col0 + t * 16 + lm;
    colc[t] = (c < CC) ? c : (CC - 1);
  }

  v8f acc[4] = {{}, {}, {}, {}};

#define ISSUE_CHUNK(c_, s_)                                                    \
  do {                                                                         \
    const uint32_t bo_ = (uint32_t)((s_) * BUF_FL * 4);                        \
    _Pragma("unroll")                                                          \
    for (int i_ = 0; i_ < 2; i_++)                                             \
      asm volatile("global_load_async_to_lds_b128 %0, %1, off"                 \
                   :: "v"(a_l[i_] + bo_),                                      \
                      "v"((unsigned long long)(uintptr_t)(a_g[i_] + (c_) * KC))\
                   : "memory");                                                \
    _Pragma("unroll")                                                          \
    for (int i_ = 0; i_ < 4; i_++)                                             \
      asm volatile("global_load_async_to_lds_b128 %0, %1, off"                 \
                   :: "v"(b_l[i_] + bo_),                                      \
                      "v"((unsigned long long)(uintptr_t)(b_g[i_] + (c_) * KC))\
                   : "memory");                                                \
  } while (0)

  ISSUE_CHUNK(0, 0);

  int buf = 0;
  for (int c = 0; c < DD / KC; c++) {
    if (c < DD / KC - 1) {
      ISSUE_CHUNK(c + 1, buf ^ 1);
      // prev chunk (6 ops) complete when <=6 remain outstanding (in-order)
      asm volatile("s_wait_asynccnt 0x6" ::: "memory");
    } else {
      asm volatile("s_wait_asynccnt 0x0" ::: "memory");
    }
    __syncthreads();                 // chunk c visible to all waves

    const float* base = smem + buf * BUF_FL;
#pragma unroll
    for (int k = 0; k < KC; k += 4) {
      v2f a = *(const v2f*)(base + aoff + k);
#pragma unroll
      for (int t = 0; t < 4; t++) {
        v2f b = *(const v2f*)(base + boff[t] + k);
        acc[t] = __builtin_amdgcn_wmma_f32_16x16x4_f32(
            false, a, false, b, (short)0, acc[t], false, false);
      }
    }
    __syncthreads();                 // all waves done with buf before reuse
    buf ^= 1;
  }
#undef ISSUE_CHUNK

  // ---- epilogue: scale by rnorm, clip, margin at label, online softmax ----
  float rn[4];
#pragma unroll
  for (int t = 0; t < 4; t++) rn[t] = rnorm[colc[t]];

  const int strip = blockIdx.x * 2 + waveN;

#pragma unroll
  for (int r = 0; r < 8; r++) {
    const int rowr = row0 + 8 * half + r;   // C-layout: vgpr r, upper half -> +8
    const int lbl  = labels[rowr];
    float vals[4];
#pragma unroll
    for (int t = 0; t < 4; t++) {
      const int col = col0 + t * 16 + lm;
      float x = acc[t][r] * rn[t];
      x = fminf(1.0f, fmaxf(-1.0f, x));
      float logit = x * SCALE_F;
      if (col == lbl) {
        float marg = cosf(acosf(x) + MARGIN_F) * SCALE_F;
        logit = marg;
        tlog[rowr] = marg;                  // exactly one writer per row globally
      }
      if (col >= CC) logit = NEG_INF;       // mask out-of-range columns
      vals[t] = logit;
    }
    float m = fmaxf(fmaxf(vals[0], vals[1]), fmaxf(vals[2], vals[3]));
#pragma unroll
    for (int mk = 8; mk >= 1; mk >>= 1) m = fmaxf(m, __shfl_xor(m, mk, 32));
    float s = 0.0f;
    if (m > NEG_INF) {                      // uniform within 16-lane group
#pragma unroll
      for (int t = 0; t < 4; t++) s += expf(vals[t] - m);
    }
#pragma unroll
    for (int mk = 8; mk >= 1; mk >>= 1) s += __shfl_xor(s, mk, 32);
    if (lm == 0) {
      pmax[(size_t)rowr * NSTRIP + strip] = m;
      psum[(size_t)rowr * NSTRIP + strip] = s;
    }
  }
}

// ---------------------------------------------------------------------------
// Kernel 3: single block; fold 1564 strip partials per row into
// loss_b = gmax + log(sum) - target_logit; mean over B -> d_out[0].
// ---------------------------------------------------------------------------
__global__ __launch_bounds__(512) void loss_reduce_kernel(
    const float* __restrict__ ws, float* __restrict__ out) {
  const float* pmaxA = ws + OFF_PMAX;
  const float* psumA = ws + OFF_PSUM;
  const float* tlogA = ws + OFF_TLOG;
  const float NEG_INF = -__builtin_inff();

  __shared__ float sacc[16];
  const int wave = threadIdx.x >> 5;
  const int lane = threadIdx.x & 31;

  float acc = 0.0f;
  for (int row = wave; row < BB; row += 16) {
    const float* pm = pmaxA + (size_t)row * NSTRIP;
    const float* ps = psumA + (size_t)row * NSTRIP;
    float m = NEG_INF;
    for (int s = lane; s < NSTRIP; s += 32) m = fmaxf(m, pm[s]);
#pragma unroll
    for (int mk = 16; mk >= 1; mk >>= 1) m = fmaxf(m, __shfl_xor(m, mk, 32));
    float sum = 0.0f;
    for (int s = lane; s < NSTRIP; s += 32) {
      float pmv = pm[s];
      float psv = ps[s];
      sum += (pmv > NEG_INF) ? psv * expf(pmv - m) : 0.0f;
    }
#pragma unroll
    for (int mk = 16; mk >= 1; mk >>= 1) sum += __shfl_xor(sum, mk, 32);
    acc += m + logf(sum) - tlogA[row];      // identical in all lanes
  }
  if (lane == 0) sacc[wave] = acc;
  __syncthreads();
  if (threadIdx.x == 0) {
    float t = 0.0f;
    for (int i = 0; i < 16; i++) t += sacc[i];
    out[0] = t / (float)BB;
  }
}

// ---------------------------------------------------------------------------
extern "C" void kernel_launch(void* const* d_in, const int* in_sizes, int n_in,
                              void* d_out, int out_size, void* d_ws, size_t ws_size,
                              hipStream_t stream) {
  const float* emb    = (const float*)d_in[0];   // [512, 256] f32
  const float* wgt    = (const float*)d_in[1];   // [100000, 256] f32
  const int*   labels = (const int*)d_in[2];     // [512] int
  float* ws  = (float*)d_ws;
  float* out = (float*)d_out;

  wnorm_kernel<<<CC / 8, 256, 0, stream>>>(wgt, ws + OFF_RNORM);

  dim3 g2(NBLK, BB / 64);   // 782 column blocks x 8 row blocks
  gemm_softmax_kernel<<<g2, 256, 0, stream>>>(emb, wgt, labels, ws);

  loss_reduce_kernel<<<1, 512, 0, stream>>>(ws, out);
}